// CodebookLayer_29772713295963
// MI455X (gfx1250) — compile-verified
//
#include <hip/hip_runtime.h>

typedef __attribute__((ext_vector_type(2))) float v2f;
typedef __attribute__((ext_vector_type(8))) float v8f;

#define DDIM   768
#define NCODES 8192
#define NTOK   8192
#define XPAD   772            // 768 + 4 -> row stride = 4 banks, conflict-free A loads
#define CPAD   19             // score-tile row stride: conflict-free scatter AND scan
#define NEGINF (-3.402823466e38f)

// ---------------- kernel 1: c2[n] = sum_d codebook[n][d]^2 ----------------
__global__ __launch_bounds__(256) void snap_c2_kernel(const float* __restrict__ cb,
                                                      float* __restrict__ c2) {
    const int wave = threadIdx.x >> 5;
    const int lane = threadIdx.x & 31;
    const int row  = blockIdx.x * 8 + wave;
    const float* p = cb + (size_t)row * DDIM;
    float s = 0.f;
    for (int j = lane; j < DDIM; j += 32) { float v = p[j]; s += v * v; }
    #pragma unroll
    for (int off = 16; off > 0; off >>= 1) s += __shfl_down(s, off, 32);
    if (lane == 0) c2[row] = s;
}

// ---------------- kernel 2: fused GEMM + top-8 + gather-average ----------------
struct SnapSMem {
    union {
        float xs[16 * XPAD];                      // x tile (phase 1)
        struct {                                  // merge bufs (phase 2, overlays xs)
            float ms[16][128];
            int   mi[16][128];
            int   fidx[16][8];
        } m;
    } u;
    float cst[8][16 * CPAD];                      // per-wave 16x16 score tile
};

__global__ __launch_bounds__(256) void snap_topk_kernel(const float* __restrict__ x,
                                                        const float* __restrict__ cb,
                                                        const float* __restrict__ c2,
                                                        float* __restrict__ out,
                                                        float* __restrict__ ids) {
    __shared__ SnapSMem sm;
    const int tid  = threadIdx.x;
    const int wave = tid >> 5;
    const int lane = tid & 31;
    const int tok0 = blockIdx.x * 16;

    // ---- stage x tile [16 x 768] into padded LDS ----
    {
        const int lr = tid >> 4, lc = tid & 15;
        const float* xrow = x + (size_t)(tok0 + lr) * DDIM;
        #pragma unroll 4
        for (int j = 0; j < 48; ++j) {
            int c = lc + j * 16;
            sm.u.xs[lr * XPAD + c] = xrow[c];
        }
    }
    __syncthreads();

    // every lane keeps a top-8 for row (lane&15) over its half of the columns
    float best[8]; int bidx[8];
    #pragma unroll
    for (int p = 0; p < 8; ++p) { best[p] = NEGINF; bidx[p] = 0; }

    const int hrow  = lane & 15;            // A row M / B col N / C col N
    const int khalf = (lane >> 4) * 2;      // A,B: lanes 0-15 -> K=0,1 ; 16-31 -> K=2,3
    const int chalf = (lane >> 4) * 8;      // scan: which 8-column half of the tile
    const int mbase = (lane >> 4) * 8;      // C: VGPR i -> row i (lo) / i+8 (hi)
    const float* ap = &sm.u.xs[hrow * XPAD + khalf];

    for (int t = 0; t < 16; ++t) {
        const int n0 = (t * 8 + wave) * 64;               // 4 tiles: cols n0..n0+63
        const float* bp = cb + (size_t)(n0 + hrow) * DDIM + khalf;

        v8f accs[4];
        #pragma unroll
        for (int q = 0; q < 4; ++q) accs[q] = (v8f){};

        #pragma unroll 4
        for (int k0 = 0; k0 < DDIM; k0 += 4) {
            v2f a;  a.x  = ap[k0];                 a.y  = ap[k0 + 1];
            v2f b0; b0.x = bp[k0];                 b0.y = bp[k0 + 1];
            v2f b1; b1.x = bp[k0 + 16 * DDIM];     b1.y = bp[k0 + 16 * DDIM + 1];
            v2f b2; b2.x = bp[k0 + 32 * DDIM];     b2.y = bp[k0 + 32 * DDIM + 1];
            v2f b3; b3.x = bp[k0 + 48 * DDIM];     b3.y = bp[k0 + 48 * DDIM + 1];
            accs[0] = __builtin_amdgcn_wmma_f32_16x16x4_f32(
                          false, a, false, b0, (short)0, accs[0], false, false);
            accs[1] = __builtin_amdgcn_wmma_f32_16x16x4_f32(
                          false, a, false, b1, (short)0, accs[1], false, false);
            accs[2] = __builtin_amdgcn_wmma_f32_16x16x4_f32(
                          false, a, false, b2, (short)0, accs[2], false, false);
            accs[3] = __builtin_amdgcn_wmma_f32_16x16x4_f32(
                          false, a, false, b3, (short)0, accs[3], false, false);
        }

        // epilogue: per sub-tile, scatter scores 2*xc - c2 to LDS, then full-wave scan.
        // (LDS ops are in-order within a wave: no barrier needed.)
        #pragma unroll
        for (int q = 0; q < 4; ++q) {
            const int nq = n0 + q * 16;
            const float c2col = c2[nq + hrow];
            #pragma unroll
            for (int i = 0; i < 8; ++i)
                sm.cst[wave][(mbase + i) * CPAD + hrow] = 2.0f * accs[q][i] - c2col;

            const float* crow = &sm.cst[wave][hrow * CPAD + chalf];
            #pragma unroll
            for (int j = 0; j < 8; ++j) {
                float cv = crow[j]; int ci = nq + chalf + j;
                if (cv > best[7]) {
                    #pragma unroll
                    for (int p = 0; p < 8; ++p) {
                        bool sw = cv > best[p];
                        float tf = best[p]; int ti = bidx[p];
                        best[p] = sw ? cv : best[p]; bidx[p] = sw ? ci : bidx[p];
                        cv = sw ? tf : cv;           ci = sw ? ti : ci;
                    }
                }
            }
        }
    }

    __syncthreads();   // xs dead; merge buffers overlay it

    {
        const int slot = wave * 16 + chalf;
        #pragma unroll
        for (int p = 0; p < 8; ++p) {
            sm.u.m.ms[hrow][slot + p] = best[p];
            sm.u.m.mi[hrow][slot + p] = bidx[p];
        }
    }
    __syncthreads();

    // ---- final top-8 of 128 per row (stable: ties -> lowest index, like lax.top_k) ----
    if (tid < 16) {
        const int row = tid;
        float fb[8]; int fi[8];
        #pragma unroll
        for (int p = 0; p < 8; ++p) { fb[p] = NEGINF; fi[p] = 0x7fffffff; }
        for (int j = 0; j < 128; ++j) {
            float cv = sm.u.m.ms[row][j]; int ci = sm.u.m.mi[row][j];
            if ((cv > fb[7]) || (cv == fb[7] && ci < fi[7])) {
                #pragma unroll
                for (int p = 0; p < 8; ++p) {
                    bool sw = (cv > fb[p]) || (cv == fb[p] && ci < fi[p]);
                    float tf = fb[p]; int ti = fi[p];
                    fb[p] = sw ? cv : fb[p]; fi[p] = sw ? ci : fi[p];
                    cv = sw ? tf : cv;       ci = sw ? ti : ci;
                }
            }
        }
        #pragma unroll
        for (int p = 0; p < 8; ++p) {
            sm.u.m.fidx[row][p] = fi[p];
            ids[(size_t)(tok0 + row) * 8 + p] = (float)fi[p];
        }
    }
    __syncthreads();

    // ---- gather 8 codebook rows per token and average (coalesced per 16-thread group) ----
    {
        const int row = tid >> 4, c0 = tid & 15;
        int idxr[8];
        #pragma unroll
        for (int p = 0; p < 8; ++p) idxr[p] = sm.u.m.fidx[row][p];
        float* orow = out + (size_t)(tok0 + row) * DDIM;
        for (int j = 0; j < 48; ++j) {
            int d = c0 + j * 16;
            float s = 0.f;
            #pragma unroll
            for (int p = 0; p < 8; ++p) s += cb[(size_t)idxr[p] * DDIM + d];
            orow[d] = s * 0.125f;
        }
    }
}

extern "C" void kernel_launch(void* const* d_in, const int* in_sizes, int n_in,
                              void* d_out, int out_size, void* d_ws, size_t ws_size,
                              hipStream_t stream) {
    (void)in_sizes; (void)n_in; (void)out_size; (void)ws_size;
    const float* x  = (const float*)d_in[0];     // [4,2048,768] f32
    const float* cb = (const float*)d_in[1];     // [8192,768]  f32
    float* out = (float*)d_out;                  // outputs: 4*2048*768 floats
    float* ids = out + (size_t)NTOK * DDIM;      // then ids: 4*2048*8 (as float)
    float* c2  = (float*)d_ws;                   // 8192 floats scratch

    snap_c2_kernel<<<NCODES / 8, 256, 0, stream>>>(cb, c2);
    snap_topk_kernel<<<NTOK / 16, 256, 0, stream>>>(x, cb, c2, out, ids);
}